// ClassCaps_86182813762001
// MI455X (gfx1250) — compile-verified
//
#include <hip/hip_runtime.h>

// ClassCaps capsule-vote kernel for gfx1250 (MI455X).
// votes[pos,i,o,m,p] = sum_n poses[pos,i,m,n] * W[i,o,n,p]  (+ coord at m<2,p=3)
//
// Transposed mapping onto V_WMMA_F32_16X16X4_F32 (K=4 == contraction depth):
//   A' (16x4 f32)  = W[i], rows = (o,p) -> 3 row-tiles cover 40 rows
//   B' (4x16 f32)  = poses, cols = (pos,m): 4 positions x 4 pose rows
//   C  (16x16 f32) = coordinate addition folded into the accumulator
//   D' rows live in the VGPR dimension -> each lane's 4 consecutive D regs are
//   4 consecutive p values => b128 NT stores, fully coalesced per position.
//
// This revision removes the last non-essential work from the hot loop:
//  - capsule index i comes from blockIdx.y (no magic division)
//  - all 4 pose B-loads prefetched in one clause before the WMMA loop
//  - coordinate values: one mod-196 + pointer select per wave, branchless
//    cached loads + v_cndmask (no exec divergence), hoisted to the prologue
//  - accumulator kept live; only elements 3 and 7 updated per group
//
// Roofline: ~0.51 GFLOP vs ~310 MB traffic -> pure store-bandwidth bound
// (~13 us floor at 23.3 TB/s). NT loads for poses (read once), NT stores for
// votes (257 MB > 192 MB L2, written once); weight/xv/yv stay cached.

typedef float v2f __attribute__((ext_vector_type(2)));
typedef float v4f __attribute__((ext_vector_type(4)));
typedef float v8f __attribute__((ext_vector_type(8)));

#define B_DIM 64
#define H_DIM 14
#define W_DIM 14
#define CIN   32
#define ADIM  4
#define COUT  10
#define HW    (H_DIM * W_DIM)             // 196
#define NPOS  (B_DIM * HW)                // 12544
#define POSE_STRIDE (CIN * ADIM * ADIM)   // 512 floats per position
#define OUT_PER_CAP (COUT * ADIM * ADIM)  // 160 floats per (pos, i)
#define GROUPS_PER_WAVE 4
#define POS_PER_WAVE (4 * GROUPS_PER_WAVE)      // 16
#define CHUNKS_PER_I (NPOS / POS_PER_WAVE)      // 784
#define WAVES_PER_BLOCK 8
#define NBLOCKS_X (CHUNKS_PER_I / WAVES_PER_BLOCK) // 98 (y-dim carries i: 32)
#define GSTRIDE (4 * CIN * OUT_PER_CAP)         // 20480 floats per group step

__global__ __launch_bounds__(256)
void caps_votes_wmma(const float* __restrict__ poses,
                     const float* __restrict__ weight,
                     const float* __restrict__ xv,
                     const float* __restrict__ yv,
                     float* __restrict__ votes) {
    const int lane  = threadIdx.x & 31;
    const int i     = blockIdx.y;                                   // capsule-in
    const int chunk = blockIdx.x * WAVES_PER_BLOCK + (threadIdx.x >> 5);
    const int posBase0 = chunk * POS_PER_WAVE;

    const int c  = lane & 15;   // A': row-in-tile / B',D': column
    const int hi = lane >> 4;   // lane half

    // ---- A' tiles = weight[i], rows (o,p), K = n (cached loads, tiny table).
    // 32-bit A layout: lane holds M = lane%16, K = vgpr + 2*(lane/16).
    v2f atile[3];
#pragma unroll
    for (int t = 0; t < 3; ++t) {
#pragma unroll
        for (int v = 0; v < 2; ++v) {
            const int row = t * 16 + c;        // (o,p) flat row
            const int k   = v + 2 * hi;        // n
            float val = 0.0f;
            if (row < COUT * ADIM) {
                const int o = row >> 2, p = row & 3;
                val = weight[i * OUT_PER_CAP + o * 16 + k * 4 + p];
            }
            atile[t][v] = val;
        }
    }

    const int pgLane   = c >> 2;  // which of the 4 positions this column covers
    const int m        = c & 3;   // pose row
    const int posLane0 = posBase0 + pgLane;

    // ---- Prefetch all pose B'-vectors (one NT b64 load clause).
    const float* bptr0 = poses + (size_t)posLane0 * POSE_STRIDE
                               + i * (ADIM * ADIM) + m * ADIM + 2 * hi;
    v2f b[GROUPS_PER_WAVE];
#pragma unroll
    for (int g = 0; g < GROUPS_PER_WAVE; ++g)
        b[g] = __builtin_nontemporal_load(
            reinterpret_cast<const v2f*>(bptr0 + g * (4 * POSE_STRIDE)));

    // ---- Coordinate values, branchless & hoisted:
    // one mod-196, pointer select once, unconditional cached loads + select.
    const float* cp = (m == 0) ? xv : yv;      // any valid table for m>=2
    int hw = posLane0 % HW;
    float cvs[GROUPS_PER_WAVE];
#pragma unroll
    for (int g = 0; g < GROUPS_PER_WAVE; ++g) {
        const float cval = cp[hw];
        cvs[g] = (m < 2) ? cval : 0.0f;
        hw += 4;
        if (hw >= HW) hw -= HW;                // pos steps by 4 => one wrap max
    }

    // One base pointer per lane; all further addressing is immediate offsets.
    float* base = votes + ((size_t)posLane0 * CIN + i) * OUT_PER_CAP
                        + m * 4 + hi * 32;

    v8f cacc = {0.0f, 0.0f, 0.0f, 0.0f, 0.0f, 0.0f, 0.0f, 0.0f};

#pragma unroll
    for (int g = 0; g < GROUPS_PER_WAVE; ++g) {
        // C carries the coordinate addition: p = elem&3 == 3, m in {0,1}.
        cacc[3] = cvs[g];
        cacc[7] = cvs[g];

        // ---- D' = A' x B' + C : three row-tiles cover o = 0..9.
        v8f d0 = __builtin_amdgcn_wmma_f32_16x16x4_f32(
            false, atile[0], false, b[g], (short)0, cacc, false, false);
        v8f d1 = __builtin_amdgcn_wmma_f32_16x16x4_f32(
            false, atile[1], false, b[g], (short)0, cacc, false, false);
        v8f d2 = __builtin_amdgcn_wmma_f32_16x16x4_f32(
            false, atile[2], false, b[g], (short)0, cacc, false, false);

        // ---- b128 NT stores: elem e of tile t -> o = t*4 + 2*hi + (e>=4),
        // p = e&3. Address = base + g*20480 + t*64 + (e>=4)*16  (floats).
        float* bg = base + g * GSTRIDE;
        __builtin_nontemporal_store(
            __builtin_shufflevector(d0, d0, 0, 1, 2, 3), (v4f*)(bg));
        __builtin_nontemporal_store(
            __builtin_shufflevector(d0, d0, 4, 5, 6, 7), (v4f*)(bg + 16));
        __builtin_nontemporal_store(
            __builtin_shufflevector(d1, d1, 0, 1, 2, 3), (v4f*)(bg + 64));
        __builtin_nontemporal_store(
            __builtin_shufflevector(d1, d1, 4, 5, 6, 7), (v4f*)(bg + 80));
        if (hi == 0) { // tile 2 rows 32..39 (o = 8,9); rows 40..47 are padding
            __builtin_nontemporal_store(
                __builtin_shufflevector(d2, d2, 0, 1, 2, 3), (v4f*)(bg + 128));
            __builtin_nontemporal_store(
                __builtin_shufflevector(d2, d2, 4, 5, 6, 7), (v4f*)(bg + 144));
        }
    }
}

extern "C" void kernel_launch(void* const* d_in, const int* in_sizes, int n_in,
                              void* d_out, int out_size, void* d_ws, size_t ws_size,
                              hipStream_t stream) {
    const float* poses  = (const float*)d_in[0];
    const float* acts   = (const float*)d_in[1];
    const float* weight = (const float*)d_in[2];
    const float* xv     = (const float*)d_in[3];
    const float* yv     = (const float*)d_in[4];
    float* out = (float*)d_out;

    dim3 grid(NBLOCKS_X, CIN, 1);   // x: position chunks, y: capsule-in index
    caps_votes_wmma<<<grid, 256, 0, stream>>>(poses, weight, xv, yv, out);

    // acts output == activations input in flat memory order -> d2d copy.
    const size_t votesElems = (size_t)NPOS * CIN * OUT_PER_CAP; // 64,225,280
    hipMemcpyAsync(out + votesElems, acts, (size_t)NPOS * CIN * sizeof(float),
                   hipMemcpyDeviceToDevice, stream);
}